// dot_attention_49065706389912
// MI455X (gfx1250) — compile-verified
//
#include <hip/hip_runtime.h>
#include <stdint.h>

typedef __attribute__((ext_vector_type(2))) float v2f;
typedef __attribute__((ext_vector_type(8))) float v8f;

#define Bdim 4
#define Hdim 8
#define Sdim 2048
#define Ddim 64
#define QT   16               // query rows per block
#define NWAVE 4               // waves per block
#define KTILES (Sdim / 16)    // 128 key tiles
#define ROWSTRIDE 2052        // padded floats per score row (bank-conflict pad)

__global__ __launch_bounds__(NWAVE * 32)
void attn_fp32_wmma_kernel(const float* __restrict__ Q,
                           const float* __restrict__ K,
                           const float* __restrict__ V,
                           const uint8_t* __restrict__ Msk,
                           float* __restrict__ Out,
                           float* __restrict__ Pr)
{
    extern __shared__ float smem[];
    float* sc   = smem;                   // 16 x ROWSTRIDE scores -> exp -> probs
    float* red  = smem + 16 * ROWSTRIDE;  // 128 partials
    float* rmax = red + 128;              // 16 row maxima
    float* rinv = rmax + 16;              // 16 reciprocal row sums

    const int tid  = threadIdx.x;
    const int lane = tid & 31;
    const int wv   = tid >> 5;
    const int r16  = lane & 15;   // row / column index within tile
    const int hi   = lane >> 4;   // lane-half selects K pair (A/B layouts)

    const int q0 = blockIdx.x * QT;
    const int h  = blockIdx.y;
    const int b  = blockIdx.z;

    const size_t bh = (size_t)(b * Hdim + h);
    const float*   Qp = Q + (bh * Sdim + q0) * Ddim;
    const float*   Kp = K + bh * Sdim * Ddim;
    const float*   Vp = V + bh * Sdim * Ddim;
    const uint8_t* Mp = Msk + ((size_t)b * Sdim + q0) * Sdim;   // mask is (B,1,S,S)
    float*         Op = Out + (bh * Sdim + q0) * Ddim;
    float*         Pp = Pr  + (bh * Sdim + q0) * Sdim;

    // ---- Q tile in WMMA A-layout: 16 chunks of 16x4, 2 VGPRs each ----
    v2f qa[16];
#pragma unroll
    for (int c = 0; c < 16; ++c)
        qa[c] = *(const v2f*)(Qp + r16 * Ddim + c * 4 + hi * 2);

    const float scale = 0.125f;  // 1/sqrt(64)

    // ================= Phase 1: S = QK^T * scale, masked, into LDS ========
    for (int kt = wv; kt < KTILES; kt += NWAVE) {
        const int k0 = kt * 16;
        v2f kb[16];
#pragma unroll
        for (int c = 0; c < 16; ++c)   // B-layout of K^T == row-major K rows
            kb[c] = *(const v2f*)(Kp + (size_t)(k0 + r16) * Ddim + c * 4 + hi * 2);
        // two independent accumulation chains -> half the XDL critical path
        v8f acc0 = {}, acc1 = {};
#pragma unroll
        for (int c = 0; c < 8; ++c) {
            acc0 = __builtin_amdgcn_wmma_f32_16x16x4_f32(
                false, qa[c],     false, kb[c],     (short)0, acc0, false, false);
            acc1 = __builtin_amdgcn_wmma_f32_16x16x4_f32(
                false, qa[c + 8], false, kb[c + 8], (short)0, acc1, false, false);
        }
        v8f acc = acc0 + acc1;
        // C-layout scatter into LDS with mask applied
#pragma unroll
        for (int i = 0; i < 8; ++i) {
            const int m = i + hi * 8;
            float v = acc[i] * scale;
            if (Mp[(size_t)m * Sdim + k0 + r16]) v = -1e9f;
            sc[m * ROWSTRIDE + k0 + r16] = v;
        }
    }
    __syncthreads();

    // ====== Softmax statistics; leaves exp(s - rowmax) in the LDS buffer ===
    {
        const int r = tid & 15, seg = tid >> 4;      // 8 segments of 256 cols
        const int c0 = seg * (Sdim / 8);
        float mx = -3.0e38f;
        for (int c = 0; c < Sdim / 8; ++c)
            mx = fmaxf(mx, sc[r * ROWSTRIDE + c0 + c]);
        red[r * 8 + seg] = mx;
        __syncthreads();
        if (tid < 16) {
            float mm = red[tid * 8];
            for (int s2 = 1; s2 < 8; ++s2) mm = fmaxf(mm, red[tid * 8 + s2]);
            rmax[tid] = mm;
        }
        __syncthreads();
        const float rm = rmax[r];
        float sum = 0.f;
        for (int c = 0; c < Sdim / 8; ++c) {
            const int idx = r * ROWSTRIDE + c0 + c;
            float e = __expf(sc[idx] - rm);
            sum += e;
            sc[idx] = e;                 // reuse: phase 2 only scales by 1/sum
        }
        red[r * 8 + seg] = sum;
        __syncthreads();
        if (tid < 16) {
            float s = 0.f;
            for (int s2 = 0; s2 < 8; ++s2) s += red[tid * 8 + s2];
            rinv[tid] = 1.0f / s;
        }
        __syncthreads();
    }

    // ======== Phase 2: P -> global (NT stream), O += P @ V ================
    v8f oacc[4];
#pragma unroll
    for (int j = 0; j < 4; ++j) oacc[j] = (v8f){};

    for (int kt = wv; kt < KTILES; kt += NWAVE) {
        const int k0 = kt * 16;
        // normalize in C-layout, stream probs to HBM, write p back to LDS
#pragma unroll
        for (int i = 0; i < 8; ++i) {
            const int m   = i + hi * 8;
            const int idx = m * ROWSTRIDE + k0 + r16;
            float p = sc[idx] * rinv[m];
            __builtin_nontemporal_store(p, &Pp[(size_t)m * Sdim + k0 + r16]);
            sc[idx] = p;
        }
        // P (A-layout from LDS) x V (B-layout from global), K-dim = 16
#pragma unroll
        for (int c = 0; c < 4; ++c) {
            v2f pa = *(const v2f*)&sc[r16 * ROWSTRIDE + k0 + c * 4 + hi * 2];
#pragma unroll
            for (int j = 0; j < 4; ++j) {   // 4 independent O accumulators
                v2f vb;
                vb.x = Vp[(size_t)(k0 + c * 4 + hi * 2    ) * Ddim + j * 16 + r16];
                vb.y = Vp[(size_t)(k0 + c * 4 + hi * 2 + 1) * Ddim + j * 16 + r16];
                oacc[j] = __builtin_amdgcn_wmma_f32_16x16x4_f32(
                    false, pa, false, vb, (short)0, oacc[j], false, false);
            }
        }
    }

    // ======== Cross-wave O reduction through LDS, write out ===============
    __syncthreads();                 // everyone done reading sc
    float* po = smem;                // reuse score buffer: 4 x 1024 floats
#pragma unroll
    for (int j = 0; j < 4; ++j)
#pragma unroll
        for (int i = 0; i < 8; ++i) {
            const int m = i + hi * 8;
            po[wv * (QT * Ddim) + m * Ddim + j * 16 + r16] = oacc[j][i];
        }
    __syncthreads();
    {
        const int e0 = tid * 8;      // 128 threads x 8 = 1024 elements
#pragma unroll
        for (int e = e0; e < e0 + 8; ++e) {
            float s = po[e] + po[1024 + e] + po[2048 + e] + po[3072 + e];
            Op[e] = s;
        }
    }
}

extern "C" void kernel_launch(void* const* d_in, const int* in_sizes, int n_in,
                              void* d_out, int out_size, void* d_ws, size_t ws_size,
                              hipStream_t stream) {
    (void)in_sizes; (void)n_in; (void)out_size; (void)d_ws; (void)ws_size;
    const float*   Q  = (const float*)d_in[0];
    const float*   K  = (const float*)d_in[1];
    const float*   V  = (const float*)d_in[2];
    const uint8_t* M  = (const uint8_t*)d_in[3];
    float* Out = (float*)d_out;
    float* Pr  = (float*)d_out + (size_t)Bdim * Hdim * Sdim * Ddim;

    const size_t shmem = (size_t)(16 * ROWSTRIDE + 128 + 16 + 16) * sizeof(float);
    static bool attr_set = false;   // idempotent device-state config (not a capture-breaking API)
    if (!attr_set) {
        hipFuncSetAttribute((const void*)attn_fp32_wmma_kernel,
                            hipFuncAttributeMaxDynamicSharedMemorySize, (int)shmem);
        attr_set = true;
    }

    dim3 grid(Sdim / QT, Hdim, Bdim);   // 128 x 8 x 4 = 4096 blocks
    dim3 block(NWAVE * 32);             // 4 waves (wave32)
    attn_fp32_wmma_kernel<<<grid, block, shmem, stream>>>(Q, K, V, M, Out, Pr);
}